// Chamfer_F1_65481071394839
// MI455X (gfx1250) — compile-verified
//
#include <hip/hip_runtime.h>

typedef __attribute__((ext_vector_type(2))) float v2f;
typedef __attribute__((ext_vector_type(8))) float v8f;

#define F1_TH 1e-4f

// ---------------------------------------------------------------------------
// Prep: build the WMMA B-fragment stream for one point set.
// For column tile t, lane layout of B (4x16 f32, K x N):
//   lanes 0-15  : (qx, qy)   for q = t*16 + lane
//   lanes 16-31 : (qz, ||q||^2)
// Stored as bs[((b*T + t)*32 + lane)] as a v2f -> inner loop is one b64 load.
// ---------------------------------------------------------------------------
__global__ __launch_bounds__(256)
void build_bfrag(const float* __restrict__ XYZ,  // [B, M, 3]
                 float* __restrict__ bs,         // [B*T*32*2] floats
                 int M)
{
  const int gid  = blockIdx.x * 256 + threadIdx.x;  // (b*T + t)*32 + lane
  const int lane = gid & 31;
  const int tb   = gid >> 5;                        // b*T + t
  const int half = lane >> 4;
  const int l16  = lane & 15;
  const int T    = M >> 4;
  const int b    = tb / T;
  const int t    = tb - b * T;
  const int q    = b * M + t * 16 + l16;
  const float qx = XYZ[3*q+0], qy = XYZ[3*q+1], qz = XYZ[3*q+2];
  const float sqq = qx*qx + qy*qy + qz*qz;
  bs[2*gid+0] = half ? qz  : qx;
  bs[2*gid+1] = half ? sqq : qy;
}

// ---------------------------------------------------------------------------
// One wave owns 32 rows (two 16-row tiles) and sweeps all M columns.
// Per 16-col tile: one v2f (b64) B-fragment load feeds TWO
// V_WMMA_F32_16X16X4_F32 ops:
//   A row (point p) = (-2px, -2py, -2pz, 1)
//   B col (point q) = ( qx,  qy,  qz, ||q||^2 )
//   C               = ||p||^2   (loop-invariant accumulator init)
//   D = A*B + C     = squared-distance tile
// The max(.,0) clamp commutes with min, so it is applied once at the end.
// blockDim = 256 (8 waves) -> 256 rows per block.
// ---------------------------------------------------------------------------
__global__ __launch_bounds__(256)
void chamfer_min_pass(const float* __restrict__ P,   // [B, N, 3] row points
                      const float* __restrict__ bs,  // B-fragment stream of Q
                      float* __restrict__ dmin,      // [B, N]
                      int N, int M)
{
  const int b    = blockIdx.y;
  const int wave = threadIdx.x >> 5;
  const int lane = threadIdx.x & 31;
  const int half = lane >> 4;
  const int l16  = lane & 15;
  const int rowBase = blockIdx.x * 256 + wave * 32;
  const int T    = M >> 4;

  const float* Pb  = P + (size_t)b * N * 3;
  const v2f*   bsb = (const v2f*)bs + (size_t)b * T * 32;

  // ---- A + C fragments for the two row tiles (one-time setup)
  v2f a0, a1;
  v8f c0, c1;
  {
    const int pr = rowBase + l16;
    const float px = Pb[3*pr+0], py = Pb[3*pr+1], pz = Pb[3*pr+2];
    const float sqp = px*px + py*py + pz*pz;
    a0.x = half ? (-2.0f*pz) : (-2.0f*px);
    a0.y = half ? 1.0f       : (-2.0f*py);
    #pragma unroll
    for (int j = 0; j < 8; ++j) c0[j] = __shfl(sqp, j + 8*half, 32);
  }
  {
    const int pr = rowBase + 16 + l16;
    const float px = Pb[3*pr+0], py = Pb[3*pr+1], pz = Pb[3*pr+2];
    const float sqp = px*px + py*py + pz*pz;
    a1.x = half ? (-2.0f*pz) : (-2.0f*px);
    a1.y = half ? 1.0f       : (-2.0f*py);
    #pragma unroll
    for (int j = 0; j < 8; ++j) c1[j] = __shfl(sqp, j + 8*half, 32);
  }

  v8f acc0, acc1;
  #pragma unroll
  for (int j = 0; j < 8; ++j) { acc0[j] = 3.4e38f; acc1[j] = 3.4e38f; }

  // ---- main sweep: 1 coalesced b64 load + 2 WMMA + 16 v_min per tile
  for (int t = 0; t < T; ++t) {
    const v2f bf = bsb[t*32 + lane];
    v8f d0 = __builtin_amdgcn_wmma_f32_16x16x4_f32(
        false, a0, false, bf, (short)0, c0, false, false);
    v8f d1 = __builtin_amdgcn_wmma_f32_16x16x4_f32(
        false, a1, false, bf, (short)0, c1, false, false);
    #pragma unroll
    for (int j = 0; j < 8; ++j) {
      acc0[j] = fminf(acc0[j], d0[j]);
      acc1[j] = fminf(acc1[j], d1[j]);
    }
  }

  // ---- min across the 16 lanes of each half (N dimension of the tile)
  #pragma unroll
  for (int off = 8; off >= 1; off >>= 1) {
    #pragma unroll
    for (int j = 0; j < 8; ++j) {
      acc0[j] = fminf(acc0[j], __shfl_xor(acc0[j], off, 32));
      acc1[j] = fminf(acc1[j], __shfl_xor(acc1[j], off, 32));
    }
  }

  // lane 0 holds rows +0..7 / +16..23, lane 16 holds rows +8..15 / +24..31
  if (l16 == 0) {
    float* o = dmin + (size_t)b * N + rowBase + 8*half;
    #pragma unroll
    for (int j = 0; j < 8; ++j) {
      o[j]      = fmaxf(acc0[j], 0.0f);
      o[16 + j] = fmaxf(acc1[j], 0.0f);
    }
  }
}

// One block per batch: cd_p, cd_t, f1 from dist1[B,N], dist2[B,M].
__global__ __launch_bounds__(256)
void chamfer_reduce(const float* __restrict__ dist1,
                    const float* __restrict__ dist2,
                    float* __restrict__ out,      // [12] = cd_p[4] | cd_t[4] | f1[4]
                    int N, int M)
{
  const int b = blockIdx.x;
  const int t = threadIdx.x;
  float s1r = 0.f, s1 = 0.f, c1 = 0.f, s2r = 0.f, s2 = 0.f, c2 = 0.f;
  for (int i = t; i < N; i += 256) {
    float d = dist1[(size_t)b*N + i];
    s1r += sqrtf(d); s1 += d; c1 += (d < F1_TH) ? 1.0f : 0.0f;
  }
  for (int i = t; i < M; i += 256) {
    float d = dist2[(size_t)b*M + i];
    s2r += sqrtf(d); s2 += d; c2 += (d < F1_TH) ? 1.0f : 0.0f;
  }
  __shared__ float sm[256];
  float vals[6] = {s1r, s1, c1, s2r, s2, c2};
  float red[6];
  #pragma unroll
  for (int k = 0; k < 6; ++k) {
    sm[t] = vals[k];
    __syncthreads();
    for (int s = 128; s > 0; s >>= 1) {
      if (t < s) sm[t] += sm[t + s];
      __syncthreads();
    }
    red[k] = sm[0];
    __syncthreads();
  }
  if (t == 0) {
    float cd_p = 0.5f * (red[0]/(float)N + red[3]/(float)M);
    float cd_t = red[1]/(float)N + red[4]/(float)M;
    float p1 = red[2]/(float)N, p2 = red[5]/(float)M;
    float den = p1 + p2;
    float f1 = (den > 0.0f) ? (2.0f*p1*p2/den) : 0.0f;
    out[b] = cd_p; out[4+b] = cd_t; out[8+b] = f1;
  }
}

extern "C" void kernel_launch(void* const* d_in, const int* in_sizes, int n_in,
                              void* d_out, int out_size, void* d_ws, size_t ws_size,
                              hipStream_t stream) {
  const float* xyz1 = (const float*)d_in[0];
  const float* xyz2 = (const float*)d_in[1];
  float* out = (float*)d_out;

  const int B = 4;
  const int N = in_sizes[0] / (B * 3);   // 8192
  const int M = in_sizes[1] / (B * 3);   // 8192
  const int T1 = M >> 4;                 // col tiles for pass 1 (Q = xyz2)
  const int T2 = N >> 4;                 // col tiles for pass 2 (Q = xyz1)

  float* dist1 = (float*)d_ws;                    // [B, N]
  float* dist2 = dist1 + (size_t)B * N;           // [B, M]
  float* bs2   = dist2 + (size_t)B * M;           // B-frag stream of xyz2
  float* bs1   = bs2   + (size_t)B * T1 * 64;     // B-frag stream of xyz1

  build_bfrag<<<dim3((B * T1 * 32) / 256), 256, 0, stream>>>(xyz2, bs2, M);
  build_bfrag<<<dim3((B * T2 * 32) / 256), 256, 0, stream>>>(xyz1, bs1, N);

  chamfer_min_pass<<<dim3(N / 256, B), 256, 0, stream>>>(xyz1, bs2, dist1, N, M);
  chamfer_min_pass<<<dim3(M / 256, B), 256, 0, stream>>>(xyz2, bs1, dist2, M, N);

  chamfer_reduce<<<dim3(B), 256, 0, stream>>>(dist1, dist2, out, N, M);
}